// DeepFeatureMatcher_13451837571313
// MI455X (gfx1250) — compile-verified
//
#include <hip/hip_runtime.h>
#include <hip/hip_bf16.h>

typedef float v2f __attribute__((ext_vector_type(2)));
typedef float v8f __attribute__((ext_vector_type(8)));

#define NPIX   12288      // 96*128 descriptors per map
#define CH     512        // channels (K of the GEMM)
#define NCHUNK 4          // j-range split for occupancy
#define JCHUNK (NPIX / NCHUNK)           // 3072
#define JTILES (JCHUNK / 16)             // 192 (processed in pairs -> 96)
#define ROWS_PER_BLK 128                 // 4 waves * 32 rows
#define LDS_STRIDE 516                   // 512 + 4 dword pad
#define RATIO_T 0.95f
#define EPS_T   1e-08f

// WMMA K-swizzle: channel c = 4s + 2h + t  ->  offset h*256 + 2s + t
// so each lane-half's K-stream is contiguous (b128 load = 2 K-steps).
__device__ __host__ inline int kswz(int c) {
    return ((c >> 1) & 1) * 256 + ((c >> 2) << 1) + (c & 1);
}

__device__ inline v8f wmma4(float ax, float ay, float bx, float by, v8f c) {
    v2f a; a[0] = ax; a[1] = ay;
    v2f b; b[0] = bx; b[1] = by;
    return __builtin_amdgcn_wmma_f32_16x16x4_f32(false, a, false, b,
                                                 (short)0, c, false, false);
}

// branchless exact top-2 update (strict > keeps lowest index on ties)
__device__ inline void upd2(float v, int j, float& t1, float& t2, int& i1) {
    t2 = fmaxf(t2, fminf(v, t1));
    i1 = (v > t1) ? j : i1;
    t1 = fmaxf(t1, v);
}

// ---------------------------------------------------------------------------
// 1) inverse L2 norm of each descriptor (column of the CHxNPIX map)
// ---------------------------------------------------------------------------
__global__ void col_inv_norm(const float* __restrict__ src, float* __restrict__ inv) {
    int p = blockIdx.x * blockDim.x + threadIdx.x;
    if (p >= NPIX) return;
    float s = 0.0f;
    #pragma unroll 8
    for (int c = 0; c < CH; ++c) {
        float x = src[(size_t)c * NPIX + p];
        s += x * x;
    }
    inv[p] = 1.0f / sqrtf(s);
}

// ---------------------------------------------------------------------------
// 2) transpose (CH x NPIX) -> per-pixel K-swizzled rows, normalization folded
// ---------------------------------------------------------------------------
__global__ void transpose_scale(const float* __restrict__ src,
                                const float* __restrict__ inv,
                                float* __restrict__ dst) {
    __shared__ float tile[32][33];
    int p0 = blockIdx.x * 32, c0 = blockIdx.y * 32;
    int tx = threadIdx.x, ty = threadIdx.y;   // block (32,8)
    #pragma unroll
    for (int k = 0; k < 4; ++k) {
        int c = c0 + ty + 8 * k;
        tile[ty + 8 * k][tx] = src[(size_t)c * NPIX + p0 + tx];
    }
    __syncthreads();
    #pragma unroll
    for (int k = 0; k < 4; ++k) {
        int p = p0 + ty + 8 * k;
        dst[(size_t)p * CH + kswz(c0 + tx)] = tile[tx][ty + 8 * k] * inv[p];
    }
}

// ---------------------------------------------------------------------------
// 3) fused GEMM + running top-2 (one direction).
//    4 waves/WG; wave owns 2 row-strips x 2 j-tiles (2x2 register tiling).
//    A block (128 x 512) staged in LDS; B streams from L2 (fully resident).
// ---------------------------------------------------------------------------
__global__ void __launch_bounds__(128)
topk_pass(const float* __restrict__ dA,   // (NPIX x CH) swizzled rows, normalized
          const float* __restrict__ dB,
          float* __restrict__ pS1, float* __restrict__ pS2, int* __restrict__ pIx) {
    extern __shared__ float lds[];        // ROWS_PER_BLK * LDS_STRIDE floats

    const int i0   = blockIdx.x * ROWS_PER_BLK;
    const int jc   = blockIdx.y;
    const int w    = threadIdx.x >> 5;    // wave id 0..3
    const int lane = threadIdx.x & 31;
    const int ln   = lane & 15;
    const int h    = lane >> 4;           // K-half selector (f32 WMMA layout)

    // stage A block into LDS (rows already K-swizzled), coalesced float4
    const float* Ablk = dA + (size_t)i0 * CH;
    for (int t = threadIdx.x; t < ROWS_PER_BLK * (CH / 4); t += 128) {
        int row = t >> 7;
        int c4  = t & 127;
        float4 v = *(const float4*)(Ablk + (size_t)row * CH + c4 * 4);
        *(float4*)(&lds[row * LDS_STRIDE + c4 * 4]) = v;
    }
    __syncthreads();

    // per-lane running top-2 per row-strip
    float t1[2][8], t2[2][8];
    int   ix[2][8];
    #pragma unroll
    for (int s = 0; s < 2; ++s)
        #pragma unroll
        for (int r = 0; r < 8; ++r) { t1[s][r] = -2.0f; t2[s][r] = -2.0f; ix[s][r] = 0; }

    const int aoff0 = (w * 32 + ln) * LDS_STRIDE + h * 256;   // 16B aligned
    const int aoff1 = aoff0 + 16 * LDS_STRIDE;

    for (int jp = 0; jp < JTILES / 2; ++jp) {
        const int j0 = jc * JCHUNK + jp * 32;
        const float* b0 = dB + (size_t)(j0 + ln) * CH + h * 256;
        const float* b1 = b0 + (size_t)16 * CH;

        v8f acc00 = {}, acc01 = {}, acc10 = {}, acc11 = {};
        #pragma unroll 2
        for (int m = 0; m < CH / 8; ++m) {          // 64 iters, 2 K-steps each
            float4 a0  = *(const float4*)(&lds[aoff0 + 4 * m]);
            float4 a1  = *(const float4*)(&lds[aoff1 + 4 * m]);
            float4 bv0 = *(const float4*)(b0 + 4 * m);
            float4 bv1 = *(const float4*)(b1 + 4 * m);
            acc00 = wmma4(a0.x, a0.y, bv0.x, bv0.y, acc00);
            acc01 = wmma4(a0.x, a0.y, bv1.x, bv1.y, acc01);
            acc10 = wmma4(a1.x, a1.y, bv0.x, bv0.y, acc10);
            acc11 = wmma4(a1.x, a1.y, bv1.x, bv1.y, acc11);
            acc00 = wmma4(a0.z, a0.w, bv0.z, bv0.w, acc00);
            acc01 = wmma4(a0.z, a0.w, bv1.z, bv1.w, acc01);
            acc10 = wmma4(a1.z, a1.w, bv0.z, bv0.w, acc10);
            acc11 = wmma4(a1.z, a1.w, bv1.z, bv1.w, acc11);
        }

        const int jc0 = j0 + ln;
        const int jc1 = j0 + 16 + ln;
        #pragma unroll
        for (int r = 0; r < 8; ++r) {
            upd2(acc00[r], jc0, t1[0][r], t2[0][r], ix[0][r]);
            upd2(acc01[r], jc1, t1[0][r], t2[0][r], ix[0][r]);
            upd2(acc10[r], jc0, t1[1][r], t2[1][r], ix[1][r]);
            upd2(acc11[r], jc1, t1[1][r], t2[1][r], ix[1][r]);
        }
    }

    // cross-lane exact top-2 merge over the 16 lanes holding each row
    #pragma unroll
    for (int s = 0; s < 2; ++s) {
        #pragma unroll
        for (int r = 0; r < 8; ++r) {
            float v1 = t1[s][r], v2 = t2[s][r];
            int   i1 = ix[s][r];
            #pragma unroll
            for (int m = 8; m >= 1; m >>= 1) {
                float o1 = __shfl_xor(v1, m, 32);
                float o2 = __shfl_xor(v2, m, 32);
                int   oi = __shfl_xor(i1, m, 32);
                bool take = (o1 > v1) || (o1 == v1 && oi < i1);
                float nv2 = take ? fmaxf(v1, o2) : fmaxf(v2, o1);
                v1 = take ? o1 : v1;
                i1 = take ? oi : i1;
                v2 = nv2;
            }
            if (ln == 0) {
                int row = i0 + w * 32 + s * 16 + r + 8 * h;
                size_t o = (size_t)jc * NPIX + row;
                pS1[o] = v1; pS2[o] = v2; pIx[o] = i1;
            }
        }
    }
}

// ---------------------------------------------------------------------------
// 4) merge the NCHUNK partial top-2 sets per row (exact)
// ---------------------------------------------------------------------------
__global__ void merge_chunks(const float* __restrict__ pS1, const float* __restrict__ pS2,
                             const int* __restrict__ pIx,
                             float* __restrict__ s1, float* __restrict__ s2,
                             int* __restrict__ nn) {
    int i = blockIdx.x * blockDim.x + threadIdx.x;
    if (i >= NPIX) return;
    float v1 = pS1[i], v2 = pS2[i];
    int   i1 = pIx[i];
    #pragma unroll
    for (int c = 1; c < NCHUNK; ++c) {
        size_t o = (size_t)c * NPIX + i;
        float o1 = pS1[o], o2 = pS2[o];
        int   oi = pIx[o];
        bool take = o1 > v1;                      // chunks ascend in j: ties keep earlier
        float nv2 = take ? fmaxf(v1, o2) : fmaxf(v2, o1);
        v1 = take ? o1 : v1;
        i1 = take ? oi : i1;
        v2 = nv2;
    }
    s1[i] = v1; s2[i] = v2; nn[i] = i1;
}

// ---------------------------------------------------------------------------
// 5) mutual-NN + ratio test; outputs (masked_sim, nn12, mask) as floats
// ---------------------------------------------------------------------------
__global__ void finalize(const float* __restrict__ s1A, const float* __restrict__ s2A,
                         const int* __restrict__ nnA,
                         const float* __restrict__ s1B, const float* __restrict__ s2B,
                         const int* __restrict__ nnB,
                         float* __restrict__ out) {
    int i = blockIdx.x * blockDim.x + threadIdx.x;
    if (i >= NPIX) return;
    float s1 = s1A[i], s2 = s2A[i];
    int   nn = nnA[i];
    float r12 = (2.0f - 2.0f * s1) / ((2.0f - 2.0f * s2) + EPS_T);
    float r21 = (2.0f - 2.0f * s1B[nn]) / ((2.0f - 2.0f * s2B[nn]) + EPS_T);
    bool mask = (nnB[nn] == i) && (r12 <= RATIO_T) && (r21 <= RATIO_T);
    out[i]            = mask ? s1 : 0.0f;
    out[NPIX + i]     = (float)nn;
    out[2 * NPIX + i] = mask ? 1.0f : 0.0f;
}

// ---------------------------------------------------------------------------
extern "C" void kernel_launch(void* const* d_in, const int* in_sizes, int n_in,
                              void* d_out, int out_size, void* d_ws, size_t ws_size,
                              hipStream_t stream) {
    const float* mapA = (const float*)d_in[0];
    const float* mapB = (const float*)d_in[1];
    float* out = (float*)d_out;

    const size_t NK = (size_t)NPIX * CH;
    float* w    = (float*)d_ws;
    float* dAn  = w;            // NPIX x CH normalized, K-swizzled rows
    float* dBn  = dAn + NK;
    float* invA = dBn + NK;
    float* invB = invA + NPIX;
    float* pS1A = invB + NPIX;                  // NCHUNK * NPIX each
    float* pS2A = pS1A + (size_t)NCHUNK * NPIX;
    int*   pIxA = (int*)(pS2A + (size_t)NCHUNK * NPIX);
    float* pS1B = (float*)(pIxA + (size_t)NCHUNK * NPIX);
    float* pS2B = pS1B + (size_t)NCHUNK * NPIX;
    int*   pIxB = (int*)(pS2B + (size_t)NCHUNK * NPIX);
    float* s1A  = (float*)(pIxB + (size_t)NCHUNK * NPIX);
    float* s2A  = s1A + NPIX;
    int*   nnA  = (int*)(s2A + NPIX);
    float* s1B  = (float*)(nnA + NPIX);
    float* s2B  = s1B + NPIX;
    int*   nnB  = (int*)(s2B + NPIX);

    // norms
    col_inv_norm<<<NPIX / 256, 256, 0, stream>>>(mapA, invA);
    col_inv_norm<<<NPIX / 256, 256, 0, stream>>>(mapB, invB);

    // transpose + normalize into WMMA-fragment-swizzled layout
    dim3 tb(32, 8), tg(NPIX / 32, CH / 32);
    transpose_scale<<<tg, tb, 0, stream>>>(mapA, invA, dAn);
    transpose_scale<<<tg, tb, 0, stream>>>(mapB, invB, dBn);

    // fused GEMM + top-2, both directions
    dim3 gg(NPIX / ROWS_PER_BLK, NCHUNK);
    size_t ldsBytes = (size_t)ROWS_PER_BLK * LDS_STRIDE * sizeof(float); // 264192 B
    topk_pass<<<gg, 128, ldsBytes, stream>>>(dAn, dBn, pS1A, pS2A, pIxA);
    topk_pass<<<gg, 128, ldsBytes, stream>>>(dBn, dAn, pS1B, pS2B, pIxB);

    // merge chunks
    merge_chunks<<<NPIX / 256, 256, 0, stream>>>(pS1A, pS2A, pIxA, s1A, s2A, nnA);
    merge_chunks<<<NPIX / 256, 256, 0, stream>>>(pS1B, pS2B, pIxB, s1B, s2B, nnB);

    // mutual NN + ratio test
    finalize<<<NPIX / 256, 256, 0, stream>>>(s1A, s2A, nnA, s1B, s2B, nnB, out);
}